// Waveform_Attention_55808805044781
// MI455X (gfx1250) — compile-verified
//
#include <hip/hip_runtime.h>
#include <cstdint>

// ---------------------------------------------------------------------------
// Waveform attention: depthwise conv1d(K=5) -> LeakyReLU -> conv1d(K=5)
//                     -> sigmoid -> x * att     (B=128, C=20, L=10000, fp32)
// Memory-bound (~205 MB @ 23.3 TB/s => ~9us floor). Strategy: stage each x
// tile into LDS once via CDNA5 async global->LDS b128 copies, run both convs
// out of LDS, stream gated output with non-temporal b128 stores.
// ---------------------------------------------------------------------------

#define AS1 __attribute__((address_space(1)))
#define AS3 __attribute__((address_space(3)))

// Native clang vectors (HIP's float4 is a struct and is rejected by
// __builtin_nontemporal_store; the async-LDS builtin wants v4i pointers).
typedef int   v4i __attribute__((vector_size(4 * sizeof(int))));
typedef float v4f __attribute__((vector_size(4 * sizeof(float))));

#if __has_builtin(__builtin_amdgcn_global_load_async_to_lds_b128)
#define USE_ASYNC_LDS 1
#endif

constexpr int   KSZ       = 5;
constexpr int   PADW      = 2;     // (K-1)/2
constexpr int   HALO      = 4;     // two stacked K=5 convs -> x halo of 4
constexpr int   TILE      = 2048;  // elements of L per workgroup tile
constexpr int   BLOCK     = 256;   // 8 wave32 waves
constexpr float NEG_SLOPE = 0.01f;

__global__ __launch_bounds__(BLOCK)
void waveform_attn_kernel(const float* __restrict__ x,
                          const float* __restrict__ w1,
                          const float* __restrict__ b1,
                          const float* __restrict__ w2,
                          const float* __restrict__ b2,
                          float* __restrict__ out,
                          int C, int L)
{
    // x tile [tile0-4, tile0+TILE+4) and h tile [tile0-2, tile0+TILE+2)
    __shared__ __align__(16) float sx[TILE + 2 * HALO];   // 2056 floats
    __shared__ __align__(16) float sh[TILE + 2 * PADW];   // 2052 floats

    // Grid: (tiles, C, B) -> channel comes straight from blockIdx.y,
    // no integer division/modulo in the kernel.
    const int c     = blockIdx.y;
    const int row   = blockIdx.z * C + c;    // b*C + c
    const int tile0 = blockIdx.x * TILE;
    const int tid   = threadIdx.x;

    const long long base = (long long)row * (long long)L;
    const float* __restrict__ xrow = x + base;
    float* __restrict__ orow = out + base;

    // Per-channel weights: uniform across the block (blockIdx-derived),
    // compiler scalarizes these into SGPRs (s_load).
    const float k10 = w1[c * KSZ + 0], k11 = w1[c * KSZ + 1],
                k12 = w1[c * KSZ + 2], k13 = w1[c * KSZ + 3],
                k14 = w1[c * KSZ + 4], bb1 = b1[c];
    const float k20 = w2[c * KSZ + 0], k21 = w2[c * KSZ + 1],
                k22 = w2[c * KSZ + 2], k23 = w2[c * KSZ + 3],
                k24 = w2[c * KSZ + 4], bb2 = b2[c];

    // ---- Stage x tile (+halo) into LDS; zero-pad outside the row. --------
    // L and TILE are multiples of 4, so every float4 chunk is 16B aligned
    // and never straddles the row boundary: chunks are fully in or fully out.
    constexpr int NCHUNK = (TILE + 2 * HALO) / 4;   // 514
    for (int i4 = tid; i4 < NCHUNK; i4 += BLOCK) {
        const int gf = tile0 - HALO + i4 * 4;       // float index within row
        if (gf >= 0 && gf < L) {
#ifdef USE_ASYNC_LDS
            __builtin_amdgcn_global_load_async_to_lds_b128(
                (AS1 v4i*)(xrow + gf), (AS3 v4i*)&sx[i4 * 4],
                /*imm offset*/ 0, /*cpol*/ 0);
#else
            *(v4f*)&sx[i4 * 4] = *(const v4f*)(xrow + gf);
#endif
        } else {
            *(v4f*)&sx[i4 * 4] = (v4f){0.f, 0.f, 0.f, 0.f};
        }
    }
#ifdef USE_ASYNC_LDS
#if __has_builtin(__builtin_amdgcn_s_wait_asynccnt)
    __builtin_amdgcn_s_wait_asynccnt(0);
#else
    asm volatile("s_wait_asynccnt 0" ::: "memory");
#endif
#endif
    __syncthreads();

    // ---- conv1 + LeakyReLU into sh (h forced to 0 outside [0,L)). --------
    constexpr int NH4 = (TILE + 2 * PADW) / 4;      // 513 (2052 = 513*4)
    for (int i4 = tid; i4 < NH4; i4 += BLOCK) {
        const int j0 = i4 * 4;
        v4f hv;
#pragma unroll
        for (int m = 0; m < 4; ++m) {
            const int j = j0 + m;                   // h index in tile
            const int p = tile0 - PADW + j;         // position in row
            float a = fmaf(k10, sx[j + 0],
                      fmaf(k11, sx[j + 1],
                      fmaf(k12, sx[j + 2],
                      fmaf(k13, sx[j + 3],
                      fmaf(k14, sx[j + 4], bb1)))));
            a = (a >= 0.f) ? a : NEG_SLOPE * a;
            hv[m] = (p >= 0 && p < L) ? a : 0.f;    // conv2 zero-padding
        }
        *(v4f*)&sh[j0] = hv;
    }
    __syncthreads();

    // ---- conv2 + sigmoid + gate; coalesced NT b128 stores. ---------------
    constexpr int NO4 = TILE / 4;                   // 512
    for (int i4 = tid; i4 < NO4; i4 += BLOCK) {
        const int j0 = i4 * 4;
        const int p0 = tile0 + j0;
        if (p0 >= L) continue;                      // whole chunk OOB (L%4==0)
        v4f ov;
#pragma unroll
        for (int m = 0; m < 4; ++m) {
            const int j = j0 + m;
            float a = fmaf(k20, sh[j + 0],
                      fmaf(k21, sh[j + 1],
                      fmaf(k22, sh[j + 2],
                      fmaf(k23, sh[j + 3],
                      fmaf(k24, sh[j + 4], bb2)))));
            const float att = 1.0f / (1.0f + __expf(-a));
            ov[m] = sx[j + HALO] * att;             // x[p] lives at sx[j+4]
        }
#if __has_builtin(__builtin_nontemporal_store)
        __builtin_nontemporal_store(ov, (v4f*)&orow[p0]);
#else
        *(v4f*)&orow[p0] = ov;
#endif
    }
}

extern "C" void kernel_launch(void* const* d_in, const int* in_sizes, int n_in,
                              void* d_out, int out_size, void* d_ws, size_t ws_size,
                              hipStream_t stream)
{
    const float* x  = (const float*)d_in[0];   // [B, C, L]
    const float* w1 = (const float*)d_in[1];   // [C, 1, 5]
    const float* b1 = (const float*)d_in[2];   // [C]
    const float* w2 = (const float*)d_in[3];   // [C, 1, 5]
    const float* b2 = (const float*)d_in[4];   // [C]
    float* out = (float*)d_out;

    const int C    = in_sizes[2];              // 20
    const int L    = 10000;                    // fixed by reference setup
    const int B    = in_sizes[0] / (C * L);    // 128

    dim3 grid((L + TILE - 1) / TILE, C, B);    // (5, 20, 128) workgroups
    waveform_attn_kernel<<<grid, BLOCK, 0, stream>>>(x, w1, b1, w2, b2, out, C, L);
}